// SPDIncreaseDim_38199439131086
// MI455X (gfx1250) — compile-verified
//
#include <hip/hip_runtime.h>

typedef __attribute__((ext_vector_type(2))) float v2f;
typedef __attribute__((ext_vector_type(4))) float v4f;
typedef __attribute__((ext_vector_type(8))) float v8f;

#define IN_DIM  64
#define OUT_DIM 96

// out[b] = add + E @ X[b] @ E^T
// One workgroup (128 threads = 4 wave32) per batch element.
// Phase 1: T = X @ E^T (64x96) -> stored transposed in LDS (T^t is 96x64).
// Phase 2: out = E @ T + add, streamed to HBM with nontemporal stores.
//
// WMMA fragment layouts (V_WMMA_F32_16X16X4_F32, wave32):
//   A (16x4, MxK): lane l holds row M = l%16; v0/v1 = K = 2*(l/16), 2*(l/16)+1
//   B (4x16, KxN): lane l holds col N = l%16; v0/v1 = K = 2*(l/16), 2*(l/16)+1
//   C/D (16x16):   lane l, vgpr r -> M = r + 8*(l/16), N = l%16
__launch_bounds__(128)
__global__ void spd_increase_dim_kernel(const float* __restrict__ X,
                                        const float* __restrict__ E,
                                        const float* __restrict__ A,
                                        float* __restrict__ out) {
  __shared__ __align__(16) float ldsX[IN_DIM * IN_DIM];    // 64x64  row-major
  __shared__ __align__(16) float ldsE[OUT_DIM * IN_DIM];   // 96x64  row-major (row = output index)
  __shared__ __align__(16) float ldsT[OUT_DIM * IN_DIM];   // T^t: 96x64, Tt[n][m] = T[m][n]

  const int b    = blockIdx.x;
  const int tid  = threadIdx.x;
  const int wave = tid >> 5;
  const int lane = tid & 31;
  const int half = lane >> 4;   // which 16-lane half of the wave
  const int lr   = lane & 15;

  const float* __restrict__ Xg = X + (size_t)b * (IN_DIM * IN_DIM);

  // ---- Phase 0: stage X (16 KiB, streamed) and E (24 KiB, reused) into LDS ----
  {
    const v4f* xs = (const v4f*)Xg;
    v4f* xd = (v4f*)ldsX;
#pragma unroll
    for (int i = 0; i < 8; ++i)                 // 1024 float4 / 128 threads
      xd[tid + 128 * i] = __builtin_nontemporal_load(&xs[tid + 128 * i]);

    const v4f* es = (const v4f*)E;
    v4f* ed = (v4f*)ldsE;
#pragma unroll
    for (int i = 0; i < 12; ++i)                // 1536 float4 / 128 threads
      ed[tid + 128 * i] = es[tid + 128 * i];
  }
  __syncthreads();

  // ---- Phase 1: T = X @ E^T. Wave w owns T rows [16w, 16w+16). 6 n-tiles as 3 pairs. ----
  {
    const int m0 = wave * 16;
#pragma unroll
    for (int np = 0; np < 3; ++np) {
      const int n0a = 32 * np;
      const int n0b = n0a + 16;
      v8f c0 = {};
      v8f c1 = {};
#pragma unroll
      for (int k0 = 0; k0 < IN_DIM; k0 += 4) {
        const int kc = k0 + 2 * half;
        v2f a  = *(const v2f*)&ldsX[(m0  + lr) * IN_DIM + kc];       // X[m][k..k+1]
        v2f b0 = *(const v2f*)&ldsE[(n0a + lr) * IN_DIM + kc];       // E^T[k][n] = E[n][k]
        v2f b1 = *(const v2f*)&ldsE[(n0b + lr) * IN_DIM + kc];
        c0 = __builtin_amdgcn_wmma_f32_16x16x4_f32(false, a, false, b0, (short)0, c0, false, false);
        c1 = __builtin_amdgcn_wmma_f32_16x16x4_f32(false, a, false, b1, (short)0, c1, false, false);
      }
      // Store D transposed: lane writes 8 contiguous floats -> 2x ds_store_b128.
      {
        const int ta = (n0a + lr) * IN_DIM + m0 + 8 * half;
        const int tb = (n0b + lr) * IN_DIM + m0 + 8 * half;
        *(v4f*)&ldsT[ta]     = __builtin_shufflevector(c0, c0, 0, 1, 2, 3);
        *(v4f*)&ldsT[ta + 4] = __builtin_shufflevector(c0, c0, 4, 5, 6, 7);
        *(v4f*)&ldsT[tb]     = __builtin_shufflevector(c1, c1, 0, 1, 2, 3);
        *(v4f*)&ldsT[tb + 4] = __builtin_shufflevector(c1, c1, 4, 5, 6, 7);
      }
    }
  }
  __syncthreads();

  // ---- Phase 2: out = E @ T + add. 36 tiles = 18 pairs round-robined over 4 waves. ----
  float* __restrict__ og = out + (size_t)b * (OUT_DIM * OUT_DIM);
  for (int p = wave; p < 18; p += 4) {
    const int mt  = p / 3;
    const int np  = p % 3;
    const int m0  = 16 * mt;
    const int n0a = 32 * np;
    const int n0b = n0a + 16;

    // C init from add (L2-resident; each half-wave reads contiguous 64B per row).
    v8f c0, c1;
#pragma unroll
    for (int r = 0; r < 8; ++r) {
      const int row = m0 + r + 8 * half;
      c0[r] = A[row * OUT_DIM + n0a + lr];
      c1[r] = A[row * OUT_DIM + n0b + lr];
    }

#pragma unroll
    for (int k0 = 0; k0 < IN_DIM; k0 += 4) {
      const int kc = k0 + 2 * half;
      v2f a  = *(const v2f*)&ldsE[(m0  + lr) * IN_DIM + kc];         // E[m][k..k+1]
      v2f b0 = *(const v2f*)&ldsT[(n0a + lr) * IN_DIM + kc];         // T[k][n] = Tt[n][k]
      v2f b1 = *(const v2f*)&ldsT[(n0b + lr) * IN_DIM + kc];
      c0 = __builtin_amdgcn_wmma_f32_16x16x4_f32(false, a, false, b0, (short)0, c0, false, false);
      c1 = __builtin_amdgcn_wmma_f32_16x16x4_f32(false, a, false, b1, (short)0, c1, false, false);
    }

    // Streaming stores: half-wave covers a contiguous 64B segment per (r, tile).
#pragma unroll
    for (int r = 0; r < 8; ++r) {
      const int row = m0 + r + 8 * half;
      __builtin_nontemporal_store(c0[r], &og[row * OUT_DIM + n0a + lr]);
      __builtin_nontemporal_store(c1[r], &og[row * OUT_DIM + n0b + lr]);
    }
  }
}

extern "C" void kernel_launch(void* const* d_in, const int* in_sizes, int n_in,
                              void* d_out, int out_size, void* d_ws, size_t ws_size,
                              hipStream_t stream) {
  const float* X = (const float*)d_in[0];   // (B, 64, 64) fp32
  const float* E = (const float*)d_in[1];   // (96, 64)    fp32
  const float* A = (const float*)d_in[2];   // (96, 96)    fp32
  float* out = (float*)d_out;               // (B, 96, 96) fp32

  const int batch = in_sizes[0] / (IN_DIM * IN_DIM);
  spd_increase_dim_kernel<<<batch, 128, 0, stream>>>(X, E, A, out);
}